// GraphNetwork_43207370997883
// MI455X (gfx1250) — compile-verified
//
#include <hip/hip_runtime.h>

// ---------------------------------------------------------------------------
// GraphNetwork (3-layer GAT, N=50000, E=400000, HID=128, HEADS=4) for MI455X.
// GEMMs on V_WMMA_F32_16X16X4_F32 with LDS-staged (transposed, padded) weight
// strips: per 4 WMMAs -> 1 global_load_b64 (A, reused x4) + 4 ds_load_b64 (B),
// all with immediate offsets (K/NC are template constants).
// ---------------------------------------------------------------------------

#define N_NODES 50000
#define N_EDGES 400000
#define HID     128
#define HEADS   4
#define NL      3

typedef float v2f __attribute__((ext_vector_type(2)));
typedef float v8f __attribute__((ext_vector_type(8)));

enum { EPI_NONE = 0, EPI_BIAS = 1, EPI_BIAS_RELU = 2, EPI_BIAS_RES = 3 };

#define KC   128          // K-chunk staged in LDS per iteration
#define KCP  (KC + 4)     // +4 floats pad: ds_load_b64 banks = (4*lane+k)%64, conflict-free

// C[M,NC] = A[M,K] @ B[K,NC] (+ bias / relu / residual).
// Block = 256 threads = 8 waves. Block tile = 128 rows x 64 cols.
// Wave w -> rows (mblk*8+w)*16..+15; all waves share the same 64-col strip,
// whose K x 64 slice of B is staged transposed in LDS (chunked by KC).
template <int K, int NC, int EPI>
__global__ __launch_bounds__(256) void gemm_wmma_f32(
    const float* __restrict__ A, const float* __restrict__ B,
    const float* __restrict__ bias, const float* __restrict__ res,
    float* __restrict__ C, int M)
{
    constexpr int NBLK   = NC / 64;   // 64-col strips
    constexpr int CHUNKS = K / KC;

    __shared__ float BT[64 * KCP];    // BT[n][k] = B[k][bn0+n]

    const int tid  = threadIdx.x;
    const int lane = tid & 31;
    const int wv   = tid >> 5;        // wave in block (0..7)
    const int l    = lane & 15;
    const int half = lane >> 4;       // wave32 halves: K pair {0,1} vs {2,3}

    const int mblk = blockIdx.x / NBLK;
    const int bn0  = (blockIdx.x % NBLK) * 64;

    const int  m0     = (mblk * 8 + wv) * 16;
    const bool active = (m0 < M);

    // A fragment source: lane l (half h) reads A[m0+l][k + 2h .. +1]
    const float* pA = A + (long)(m0 + l) * K + 2 * half;
    // B staging source column base
    const int sn = tid & 63;          // column handled by this thread
    const int sk = tid >> 6;          // starting k row (0..3)

    v8f acc[4];
#pragma unroll
    for (int j = 0; j < 4; ++j) acc[j] = v8f{0.f,0.f,0.f,0.f,0.f,0.f,0.f,0.f};

#pragma unroll
    for (int chunk = 0; chunk < CHUNKS; ++chunk) {
        const int kbase = chunk * KC;
        __syncthreads();              // previous chunk's LDS reads done
        // stage B[kbase..kbase+KC][bn0..bn0+63] transposed into LDS
#pragma unroll
        for (int kk = sk; kk < KC; kk += 4)
            BT[sn * KCP + kk] = B[(long)(kbase + kk) * NC + bn0 + sn];
        __syncthreads();

        if (active) {
            if (chunk + 1 < CHUNKS)   // gfx1250 global_prefetch of next A chunk
                __builtin_prefetch(pA + kbase + KC, 0, 3);
#pragma unroll
            for (int k = 0; k < KC; k += 4) {
                v2f a = *(const v2f*)(pA + kbase + k);   // immediate offset
#pragma unroll
                for (int j = 0; j < 4; ++j) {
                    v2f b = *(const v2f*)(&BT[(j * 16 + l) * KCP + k + 2 * half]);
                    acc[j] = __builtin_amdgcn_wmma_f32_16x16x4_f32(
                        false, a, false, b, (short)0, acc[j], false, false);
                }
            }
        }
    }

    if (!active) return;
#pragma unroll
    for (int j = 0; j < 4; ++j) {
        const int ncol = bn0 + j * 16 + l;
        float bv = 0.f;
        if (EPI != EPI_NONE) bv = bias[ncol];
#pragma unroll
        for (int jj = 0; jj < 8; ++jj) {
            const int  r   = m0 + jj + 8 * half;   // D vgpr jj: rows m0+jj / +8
            const long idx = (long)r * NC + ncol;
            float v = acc[j][jj];
            if (EPI != EPI_NONE)      v += bv;
            if (EPI == EPI_BIAS_RELU) v = v > 0.f ? v : 0.f;
            if (EPI == EPI_BIAS_RES)  v += res[idx];
            C[idx] = v;
        }
    }
}

// s_src[n,h] = dot(hW[n, h*128:+128], a_src[h]); same for s_dst.
// One wave per (node, head); shfl_xor wave32 reduction.
__global__ __launch_bounds__(256) void attn_logits(
    const float* __restrict__ hW, const float* __restrict__ a_src,
    const float* __restrict__ a_dst, float* __restrict__ ssrc,
    float* __restrict__ sdst)
{
    const int lane = threadIdx.x & 31;
    long wid  = (long)blockIdx.x * 8 + (threadIdx.x >> 5);
    int  node = (int)(wid >> 2);
    int  head = (int)(wid & 3);
    if (node >= N_NODES) return;

    const float* hp = hW + (long)node * (HEADS * HID) + head * HID + lane;
    const float* as = a_src + head * HID + lane;
    const float* ad = a_dst + head * HID + lane;
    float ss = 0.f, sd = 0.f;
#pragma unroll
    for (int i = 0; i < 4; ++i) {
        float v = hp[i * 32];
        ss += v * as[i * 32];
        sd += v * ad[i * 32];
    }
#pragma unroll
    for (int off = 16; off; off >>= 1) {
        ss += __shfl_xor(ss, off, 32);
        sd += __shfl_xor(sd, off, 32);
    }
    if (lane == 0) {
        ssrc[node * HEADS + head] = ss;
        sdst[node * HEADS + head] = sd;
    }
}

// Pass 1: eexp = exp(leakyrelu(s_src[row]+s_dst[col])); e_sum[col] += eexp.
__global__ __launch_bounds__(256) void edge_softmax_num(
    const int* __restrict__ rows, const int* __restrict__ cols,
    const float* __restrict__ ssrc, const float* __restrict__ sdst,
    float* __restrict__ eexp, float* __restrict__ esum)
{
    int e = blockIdx.x * 256 + threadIdx.x;
    if (e >= N_EDGES) return;
    int r = rows[e], c = cols[e];
    float4 s = *(const float4*)(ssrc + (long)r * HEADS);
    float4 d = *(const float4*)(sdst + (long)c * HEADS);
    const float* sp = (const float*)&s;
    const float* dp = (const float*)&d;
    float4 o;
    float* op = (float*)&o;
#pragma unroll
    for (int h = 0; h < HEADS; ++h) {
        float v = sp[h] + dp[h];
        v = v > 0.f ? v : 0.2f * v;      // LeakyReLU(0.2)
        v = expf(v);                     // raw exp (matches reference)
        op[h] = v;
        atomicAdd(esum + (long)c * HEADS + h, v);
    }
    *(float4*)(eexp + (long)e * HEADS) = o;
}

// Pass 2: agg[col] += hW[row] * alpha; alpha = eexp / (e_sum[col] + 1e-8).
// One wave per (edge, head): float4 gather + 4 global_atomic_add_f32.
__global__ __launch_bounds__(256) void edge_aggregate(
    const int* __restrict__ rows, const int* __restrict__ cols,
    const float* __restrict__ hW, const float* __restrict__ eexp,
    const float* __restrict__ esum, float* __restrict__ agg)
{
    const int lane = threadIdx.x & 31;
    long wid  = (long)blockIdx.x * 8 + (threadIdx.x >> 5);
    long e    = wid >> 2;
    int  head = (int)(wid & 3);
    if (e >= N_EDGES) return;
    int r = rows[e], c = cols[e];
    float alpha = eexp[e * HEADS + head] /
                  (esum[(long)c * HEADS + head] + 1e-8f);
    const float* src = hW  + (long)r * (HEADS * HID) + head * HID + lane * 4;
    float*       dst = agg + (long)c * (HEADS * HID) + head * HID + lane * 4;
    float4 v = *(const float4*)src;
    atomicAdd(dst + 0, v.x * alpha);
    atomicAdd(dst + 1, v.y * alpha);
    atomicAdd(dst + 2, v.z * alpha);
    atomicAdd(dst + 3, v.w * alpha);
}

__global__ void zero_f32(float* __restrict__ p, long n)
{
    long i      = (long)blockIdx.x * blockDim.x + threadIdx.x;
    long stride = (long)gridDim.x * blockDim.x;
    for (; i < n; i += stride) p[i] = 0.f;
}

// graph_embedding: per-block partial column sums -> pre-scaled atomics.
__global__ __launch_bounds__(128) void col_mean(
    const float* __restrict__ x, float* __restrict__ mean, int rows_per_blk)
{
    int  c  = threadIdx.x;
    long r0 = (long)blockIdx.x * rows_per_blk;
    long r1 = r0 + rows_per_blk;
    if (r1 > N_NODES) r1 = N_NODES;
    float s = 0.f;
    for (long r = r0; r < r1; ++r) s += x[r * HID + c];
    atomicAdd(mean + c, s * (1.0f / (float)N_NODES));
}

extern "C" void kernel_launch(void* const* d_in, const int* in_sizes, int n_in,
                              void* d_out, int out_size, void* d_ws, size_t ws_size,
                              hipStream_t stream)
{
    const float* X      = (const float*)d_in[0];
    const int*   ei     = (const int*)d_in[1];
    const float* enc_w1 = (const float*)d_in[2];
    const float* enc_b1 = (const float*)d_in[3];
    const float* enc_w2 = (const float*)d_in[4];
    const float* enc_b2 = (const float*)d_in[5];
    const float* gat_W  = (const float*)d_in[6];   // [L,128,512]
    const float* gat_as = (const float*)d_in[7];   // [L,4,128]
    const float* gat_ad = (const float*)d_in[8];   // [L,4,128]
    const float* gat_ow = (const float*)d_in[9];   // [L,512,128]
    const float* gat_ob = (const float*)d_in[10];  // [L,128]
    const float* out_w1 = (const float*)d_in[11];
    const float* out_b1 = (const float*)d_in[12];
    const float* out_w2 = (const float*)d_in[13];
    const float* out_b2 = (const float*)d_in[14];

    const int* rows = ei;             // edge_index[0]
    const int* cols = ei + N_EDGES;   // edge_index[1]

    // Workspace layout (floats): ~265 MB total
    float* ws   = (float*)d_ws;
    float* h    = ws;                                  // N*128
    float* tmp  = h    + (long)N_NODES * HID;          // N*128
    float* hW   = tmp  + (long)N_NODES * HID;          // N*512
    float* agg  = hW   + (long)N_NODES * HEADS * HID;  // N*512
    float* ssrc = agg  + (long)N_NODES * HEADS * HID;  // N*4
    float* sdst = ssrc + (long)N_NODES * HEADS;        // N*4
    float* esum = sdst + (long)N_NODES * HEADS;        // N*4
    float* eexp = esum + (long)N_NODES * HEADS;        // E*4

    const int MBLKS = (N_NODES / 16 + 7) / 8;          // 391 row-blocks of 128
    const int gb128 = MBLKS * (128 / 64);              // 782
    const int gb512 = MBLKS * (512 / 64);              // 3128
    const int logit_blocks = (int)(((long)N_NODES * HEADS + 7) / 8);
    const int agg_blocks   = (int)(((long)N_EDGES * HEADS + 7) / 8);

    // node encoder: Linear-ReLU-Linear
    gemm_wmma_f32<128, 128, EPI_BIAS_RELU><<<gb128, 256, 0, stream>>>(
        X, enc_w1, enc_b1, nullptr, tmp, N_NODES);
    gemm_wmma_f32<128, 128, EPI_BIAS><<<gb128, 256, 0, stream>>>(
        tmp, enc_w2, enc_b2, nullptr, h, N_NODES);

    for (int i = 0; i < NL; ++i) {
        // hW = h @ W_i   (N x 512)
        gemm_wmma_f32<128, 512, EPI_NONE><<<gb512, 256, 0, stream>>>(
            h, gat_W + (long)i * 128 * 512, nullptr, nullptr, hW, N_NODES);
        // per-node attention logits
        attn_logits<<<logit_blocks, 256, 0, stream>>>(
            hW, gat_as + (long)i * HEADS * HID,
            gat_ad + (long)i * HEADS * HID, ssrc, sdst);
        // edge softmax numerators + segment sums
        zero_f32<<<256, 256, 0, stream>>>(esum, (long)N_NODES * HEADS);
        edge_softmax_num<<<(N_EDGES + 255) / 256, 256, 0, stream>>>(
            rows, cols, ssrc, sdst, eexp, esum);
        // weighted message aggregation
        zero_f32<<<4096, 256, 0, stream>>>(agg, (long)N_NODES * HEADS * HID);
        edge_aggregate<<<agg_blocks, 256, 0, stream>>>(
            rows, cols, hW, eexp, esum, agg);
        // out_proj + residual: h = h + agg @ ow_i + ob_i
        gemm_wmma_f32<512, 128, EPI_BIAS_RES><<<gb128, 256, 0, stream>>>(
            agg, gat_ow + (long)i * 512 * 128, gat_ob + (long)i * 128,
            h, h, N_NODES);
    }

    // output MLP -> d_out[0 : N*128], then mean -> d_out[N*128 : +128]
    float* hout  = (float*)d_out;
    float* gmean = hout + (long)N_NODES * HID;
    gemm_wmma_f32<128, 128, EPI_BIAS_RELU><<<gb128, 256, 0, stream>>>(
        h, out_w1, out_b1, nullptr, tmp, N_NODES);
    gemm_wmma_f32<128, 128, EPI_BIAS><<<gb128, 256, 0, stream>>>(
        tmp, out_w2, out_b2, nullptr, hout, N_NODES);

    zero_f32<<<1, 128, 0, stream>>>(gmean, HID);
    const int rows_per_blk = 200;
    col_mean<<<(N_NODES + rows_per_blk - 1) / rows_per_blk, 128, 0, stream>>>(
        hout, gmean, rows_per_blk);
}